// ViscoplasticMaterialModel_8650064134695
// MI455X (gfx1250) — compile-verified
//
#include <hip/hip_runtime.h>
#include <hip/hip_bf16.h>
#include <math.h>

// ---------------------------------------------------------------------------
// Viscoplastic material model (PICNN energy/dissipation + FNO encoders).
//
// MI455X strategy (compile-only reasoning):
//  * ~85 GFLOP fp32, tiny weights, 960 *serial* solver iterations
//    -> latency bound, not HBM bound (23.3 TB/s): keep the whole solver
//       resident; one wave32 owns 16 batch rows (== WMMA M=16 layout).
//  * All GEMMs via V_WMMA_F32_16X16X4_F32 (fp32 needed for solver accuracy).
//  * FNO rfft/irfft recast as dense DFT matmuls (only 16 modes kept) -> WMMA,
//    with a single LDS twiddle table shared by fwd/inv transforms (the c_m/L
//    scale is folded into the spectral coefficients, so the inverse pass is a
//    pure transposed table read).
//  * Reference's convergence scalars (err/cnt/active) are BATCH-GLOBAL:
//    deterministic Q24 fixed-point atomic reduction + slot-based grid
//    barrier (s_sleep spin), counters reset by hipMemsetAsync per launch.
// ---------------------------------------------------------------------------

typedef float  v2f __attribute__((ext_vector_type(2)));
typedef float  v8f __attribute__((ext_vector_type(8)));

#define NB      2048      // batch
#define NT      32        // time steps
#define NITER   30
#define NIVD    8
#define TOLF    1e-3f
#define LRF     1e-3f
#define DT_INV  100.0f
#define PI2     6.28318530717958647692f

#define LDX     44        // padded stride for 16x40 x-matrix
#define LD64    68        // padded stride for 16x64 / 64x64 LDS matrices
#define LPAD    132       // padded stride for 32x128 FNO field
#define LDT     17        // padded stride for 128x16 twiddle table

// ---------------- small math helpers ----------------
__device__ __forceinline__ float sp_(float x)  { return (x > 20.f) ? x : log1pf(__expf(x)); }
__device__ __forceinline__ float sig_(float x) { return 1.f / (1.f + __expf(-x)); }
__device__ __forceinline__ float gelu_(float x) {
    float x3 = x * x * x;
    return 0.5f * x * (1.f + tanhf(0.7978845608028654f * (x + 0.044715f * x3)));
}
__device__ __forceinline__ v8f vzero() {
    v8f z;
#pragma unroll
    for (int i = 0; i < 8; ++i) z[i] = 0.f;
    return z;
}
__device__ __forceinline__ v8f wmma4(v2f a, v2f b, v8f c) {
    // D = A(16x4,f32) * B(4x16,f32) + C(16x16,f32)
    return __builtin_amdgcn_wmma_f32_16x16x4_f32(false, a, false, b, (short)0, c, false, false);
}

// Generic 16xN tile matmul: acc += A(16xK) * B(KxN-tile), K multiple of 4.
// A fetch af(m,k); B fetch bf(k,n).  ISA VGPR layout for V_WMMA_F32_16X16X4_F32:
//   A: lane<16 holds K=0,1 ; lane>=16 holds K=2,3  (per 4-wide K step)
//   B: V0 = rows {k, k+2}, V1 = rows {k+1, k+3} split across lane halves
//   D: vgpr r -> row r (lanes 0-15) / row r+8 (lanes 16-31)
template <typename AF, typename BF>
__device__ __forceinline__ v8f mm16g(int K, int lane, AF af, BF bf, v8f acc) {
    const int half = lane >> 4, mr = lane & 15, nc = lane & 15;
    for (int k = 0; k < K; k += 4) {
        const int ka = k + 2 * half;
        v2f a, b;
        a.x = af(mr, ka);     a.y = af(mr, ka + 1);
        b.x = bf(ka, nc);     b.y = bf(ka + 1, nc);
        acc = wmma4(a, b, acc);
    }
    return acc;
}
template <typename SF>
__device__ __forceinline__ void st16(v8f c, int lane, SF sf) {
    const int half = lane >> 4, nc = lane & 15;
#pragma unroll
    for (int r = 0; r < 8; ++r) sf(r + 8 * half, nc, c[r]);
}

// ---------------- parameter bundles (sorted-key order inside dicts) --------
struct PicnnP {
    const float *Wu0, *Wu2, *Wx1, *Wy1, *Wy2, *Wyu2, *Wyx1, *Wz2, *Wzu2;
    const float *b1, *b2, *bu0, *byu2, *byx1, *bzu2;
};
struct FnoP {
    const float *fc1_b, *fc1_w, *fc2_b, *fc2_w, *lift_b, *lift_w;
    const float *pb0, *pb1, *pb2, *pw0, *pw1, *pw2;
    const float *swim0, *swim1, *swim2, *swre0, *swre1, *swre2;
};

// ===========================================================================
// FNO1d encoder.  One block (2 waves) per batch row; rfft/irfft as WMMA
// matmuls against an LDS twiddle table, spectral 1x32x32 per-mode mix in VALU.
// ===========================================================================
__global__ __launch_bounds__(64) void fno_kernel(
    const float* __restrict__ in0, const float* __restrict__ in1,
    const float* __restrict__ in2, int n_ch, FnoP P, float* __restrict__ mf)
{
    __shared__ float hA[32 * LPAD], hB[32 * LPAD];
    __shared__ float Cre[128 * LDT], Cim[128 * LDT];   // cos(th), -sin(th)
    __shared__ float xfre[32 * 16], xfim[32 * 16];
    __shared__ float omre[32 * 16], omim[32 * 16];
    __shared__ float pw_s[32 * 33];
    __shared__ float gpool[32];
    __shared__ float t1[64];

    const int tid = threadIdx.x;
    const int lane = tid & 31, wid = tid >> 5;
    const int b = blockIdx.x;

    // ---- twiddle table, built once per block: th = 2*pi*l*m/128
    for (int idx = tid; idx < 128 * 16; idx += 64) {
        int l = idx >> 4, m = idx & 15;
        float th = PI2 * (float)(l * m) * (1.f / 128.f);
        Cre[l * LDT + m] = __cosf(th);
        Cim[l * LDT + m] = -__sinf(th);
    }

    // ---- lift: h[w][l] = sum_d in[d][l] * lift_w[d][w] + lift_b[w]
    for (int idx = tid; idx < 32 * 128; idx += 64) {
        int w = idx >> 7, l = idx & 127;
        float s = P.lift_b[w];
#pragma unroll
        for (int d = 0; d < 3; ++d) {
            const float* p = (d == 0) ? in0 : (d == 1) ? in1 : in2;
            if (d < n_ch) s += p[b * 128 + l] * P.lift_w[d * 32 + w];
        }
        hA[w * LPAD + l] = s;
    }
    __syncthreads();

    float* cur = hA;
    float* nxt = hB;
#pragma unroll
    for (int layer = 0; layer < 3; ++layer) {
        const float* wr_ = (layer == 0) ? P.swre0 : (layer == 1) ? P.swre1 : P.swre2;
        const float* wi_ = (layer == 0) ? P.swim0 : (layer == 1) ? P.swim1 : P.swim2;
        const float* pwp = (layer == 0) ? P.pw0   : (layer == 1) ? P.pw1   : P.pw2;
        const float* pbp = (layer == 0) ? P.pb0   : (layer == 1) ? P.pb1   : P.pb2;

        // ---- forward DFT (truncated to 16 modes): xf = h(32x128) @ C(128x16)
        for (int jj = 0; jj < 2; ++jj) {
            int job = wid * 2 + jj;           // 4 jobs over 2 waves
            int mt = job >> 1, part = job & 1;
            if (part == 0) {
                v8f c0 = mm16g(128, lane,
                    [&](int m, int k) { return cur[(mt * 16 + m) * LPAD + k]; },
                    [&](int k, int n) { return Cre[k * LDT + n]; }, vzero());
                st16(c0, lane, [&](int r, int n2, float v) { xfre[(mt * 16 + r) * 16 + n2] = v; });
            } else {
                v8f c0 = mm16g(128, lane,
                    [&](int m, int k) { return cur[(mt * 16 + m) * LPAD + k]; },
                    [&](int k, int n) { return Cim[k * LDT + n]; }, vzero());
                st16(c0, lane, [&](int r, int n2, float v) { xfim[(mt * 16 + r) * 16 + n2] = v; });
            }
        }
        __syncthreads();

        // ---- spectral mix: om[o][m] = sum_i xf[i][m] * sw[i][o][m] (complex)
        //      fold the irfft scale c_m/L into om here, so the inverse pass
        //      below is a pure transposed read of the same twiddle table.
        for (int idx = tid; idx < 512; idx += 64) {
            int o = idx >> 4, m = idx & 15;
            float ar = 0.f, ai = 0.f;
            for (int i = 0; i < 32; ++i) {
                float xr = xfre[i * 16 + m], xi2 = xfim[i * 16 + m];
                float wr = wr_[(i * 32 + o) * 16 + m], wi = wi_[(i * 32 + o) * 16 + m];
                ar += xr * wr - xi2 * wi;
                ai += xr * wi + xi2 * wr;
            }
            float sc = ((m == 0) ? 1.f : 2.f) * (1.f / 128.f);
            omre[o * 16 + m] = ar * sc;
            omim[o * 16 + m] = ai * sc;
        }
        for (int idx = tid; idx < 1024; idx += 64)
            pw_s[(idx >> 5) * 33 + (idx & 31)] = pwp[idx];
        __syncthreads();

        // ---- inverse DFT: hs[o][l] = sum_m omre*cos(th) - omim*sin(th)
        //      (Cim already holds -sin, and om carries c_m/L)
        for (int jj = 0; jj < 8; ++jj) {
            int job = wid * 8 + jj;           // 16 jobs over 2 waves
            int mt = job >> 3, nt = job & 7;
            v8f c0 = mm16g(16, lane,
                [&](int m, int k) { return omre[(mt * 16 + m) * 16 + k]; },
                [&](int k, int n) { return Cre[(nt * 16 + n) * LDT + k]; }, vzero());
            c0 = mm16g(16, lane,
                [&](int m, int k) { return omim[(mt * 16 + m) * 16 + k]; },
                [&](int k, int n) { return Cim[(nt * 16 + n) * LDT + k]; }, c0);
            st16(c0, lane, [&](int r, int n2, float v) {
                nxt[(mt * 16 + r) * LPAD + nt * 16 + n2] = v;
            });
        }
        __syncthreads();

        // ---- pointwise conv (pw^T @ h) + spectral branch + GELU -> nxt
        for (int jj = 0; jj < 8; ++jj) {
            int job = wid * 8 + jj;
            int mt = job >> 3, nt = job & 7;
            v8f c0 = mm16g(32, lane,
                [&](int m, int k) { return pw_s[k * 33 + mt * 16 + m]; },
                [&](int k, int n) { return cur[k * LPAD + nt * 16 + n]; }, vzero());
            st16(c0, lane, [&](int r, int n2, float v) {
                int o = mt * 16 + r, l = nt * 16 + n2;
                float val = nxt[o * LPAD + l] + v + pbp[o];
                nxt[o * LPAD + l] = gelu_(val);
            });
        }
        __syncthreads();
        float* tp = cur; cur = nxt; nxt = tp;
    }

    // ---- global pool + MLP head
    if (tid < 32) {
        float s = 0.f;
        for (int l = 0; l < 128; ++l) s += cur[tid * LPAD + l];
        gpool[tid] = s * (1.f / 128.f);
    }
    __syncthreads();
    {
        float s = P.fc1_b[tid];
        for (int w = 0; w < 32; ++w) s += gpool[w] * P.fc1_w[w * 64 + tid];
        t1[tid] = gelu_(s);
    }
    __syncthreads();
    if (tid < 32) {
        float s = P.fc2_b[tid];
        for (int j = 0; j < 64; ++j) s += t1[j] * P.fc2_w[j * 32 + tid];
        mf[b * 32 + tid] = s;
    }
}

// ===========================================================================
// dd-PICNN context precompute: its x-input (mf2) is constant per call, so its
// gate tensors are fixed: a_x, gk = relu(u1@Wzu2+b)*sp(Wz2), wxc = x@Wx1+b1,
// r_j = (u1@Wyu2+b)*Wy2.  Pure VALU, one thread per batch row.
// ===========================================================================
__global__ __launch_bounds__(128) void dd_precompute_kernel(
    const float* __restrict__ mf2, PicnnP dp,
    float* __restrict__ Axg, float* __restrict__ gkg,
    float* __restrict__ wxg, float* __restrict__ rjg)
{
    int b = blockIdx.x * 128 + threadIdx.x;
    if (b >= NB) return;
    float x[32];
    for (int i = 0; i < 32; ++i) x[i] = mf2[b * 32 + i];
    float u1[64];
    for (int k = 0; k < 64; ++k) {
        float s = dp.bu0[k];
        for (int i = 0; i < 32; ++i) s += x[i] * dp.Wu0[i * 64 + k];
        u1[k] = sp_(s);
    }
    for (int k = 0; k < 64; ++k) {
        float s = dp.bzu2[k];
        for (int i = 0; i < 64; ++i) s += u1[i] * dp.Wzu2[i * 64 + k];
        float zg = (s > 0.f) ? s : 0.f;
        gkg[b * 64 + k] = zg * sp_(dp.Wz2[k]);
    }
    for (int j = 0; j < 8; ++j) {
        float s = dp.byu2[j];
        for (int i = 0; i < 64; ++i) s += u1[i] * dp.Wyu2[i * 8 + j];
        rjg[b * 8 + j] = s * dp.Wy2[j];
        float a = dp.byx1[j];
        for (int i = 0; i < 32; ++i) a += x[i] * dp.Wyx1[i * 8 + j];
        Axg[b * 8 + j] = a;
    }
    for (int k = 0; k < 64; ++k) {
        float s = dp.b1[k];
        for (int i = 0; i < 32; ++i) s += x[i] * dp.Wx1[i * 64 + k];
        wxg[b * 64 + k] = s;
    }
}

// ===========================================================================
// Persistent implicit-solver kernel.  128 single-wave blocks, each owning a
// 16-row batch tile.  All PICNN fwd/bwd GEMMs are V_WMMA_F32_16X16X4_F32.
// Global (batch-wide) err/active scalars via deterministic Q24 integer
// atomics + slot-per-iteration grid barrier.
// ===========================================================================
__global__ __launch_bounds__(32) void time_loop_kernel(
    const float* __restrict__ e, const float* __restrict__ mf1,
    PicnnP ep, const float* __restrict__ Wy1d_g,
    const float* __restrict__ Axg, const float* __restrict__ gkg,
    const float* __restrict__ wxg, const float* __restrict__ rjg,
    unsigned long long* __restrict__ acc, unsigned* __restrict__ bar,
    float* __restrict__ stress, float* __restrict__ xiout)
{
    __shared__ float xs[16 * LDX];                         // [xi(8) | mf1(32) | pad]
    __shared__ float Ab[16 * LD64], Bb[16 * LD64];         // z1pre/dz, zg/dzg
    __shared__ float Ub[16 * LD64], Sb[16 * LD64];         // u1/h1, sig(u)/es
    __shared__ float Wzu2s[64 * LD64];
    __shared__ float Wy1ds[16 * LD64];                     // rows 8..15 zero
    __shared__ float gks[16 * 64], wxcs[16 * 64];
    __shared__ float gxa[16 * 17], gxb[16 * 17], ddx[16 * 17];
    __shared__ float sqs[16 * 12];
    __shared__ float Wy1row[64], wzv[64], Wyu2v[64], Wu2v[64];
    __shared__ float bu0v[64], b1v[64], bzu2v[64], Wyx1v[40];
    __shared__ float Axs[16 * 8], rjs[16 * 8];
    __shared__ float xiw[16 * 8], xipv[16 * 8], xip2[16 * 8];
    __shared__ float y_s[16], t_s[16], S_s[16], yg_s[16], gu_s[16], seq_s[16], psum[16];
    __shared__ float scal[4];
    __shared__ float errsh;

    const int lane = threadIdx.x;
    const int base = blockIdx.x * 16;
    const float* Wu0g = ep.Wu0;
    const float* Wx1g = ep.Wx1;
    __builtin_prefetch(Wu0g, 0, 1);
    __builtin_prefetch(Wx1g, 0, 1);
    __builtin_prefetch(ep.Wzu2, 0, 1);

    // ---- one-time staging -------------------------------------------------
    for (int i = lane; i < 64 * 64; i += 32)
        Wzu2s[(i >> 6) * LD64 + (i & 63)] = ep.Wzu2[i];
    for (int i = lane; i < 16 * 64; i += 32) {
        int r = i >> 6, c = i & 63;
        Wy1ds[r * LD64 + c] = (r < 8) ? Wy1d_g[r * 64 + c] : 0.f;
    }
    for (int i = lane; i < 64; i += 32) {
        Wy1row[i] = ep.Wy1[i];
        wzv[i]    = sp_(ep.Wz2[i]);
        Wyu2v[i]  = ep.Wyu2[i];
        Wu2v[i]   = ep.Wu2[i];
        bu0v[i]   = ep.bu0[i];
        b1v[i]    = ep.b1[i];
        bzu2v[i]  = ep.bzu2[i];
    }
    for (int i = lane; i < 40; i += 32) Wyx1v[i] = ep.Wyx1[i];
    if (lane == 0) { scal[0] = ep.byx1[0]; scal[1] = ep.byu2[0]; scal[2] = ep.Wy2[0]; }
    for (int i = lane; i < 16 * 32; i += 32) {
        int m = i >> 5, j = i & 31;
        xs[m * LDX + 8 + j] = mf1[(base + m) * 32 + j];
    }
    for (int i = lane; i < 16 * 4; i += 32) xs[(i >> 2) * LDX + 40 + (i & 3)] = 0.f;
    for (int i = lane; i < 16 * 8; i += 32) {
        int m = i >> 3, j = i & 7;
        Axs[i] = Axg[(base + m) * 8 + j];
        rjs[i] = rjg[(base + m) * 8 + j];
        xipv[i] = 0.f;
        xip2[i] = 0.f;
    }
    for (int i = lane; i < 16 * 64; i += 32) {
        int m = i >> 6, k = i & 63;
        gks[i]  = gkg[(base + m) * 64 + k];
        wxcs[i] = wxg[(base + m) * 64 + k];
    }
    const float byx1s = scal[0], byu2s = scal[1], Wy2s = scal[2];

    // ---- time stepping ----------------------------------------------------
    for (int t = 0; t < NT; ++t) {
        if (lane < 16) {
            y_s[lane] = e[(base + lane) * NT + t];
            seq_s[lane] = 0.f;
#pragma unroll
            for (int j = 0; j < NIVD; ++j) {
                float x0 = (t > 2) ? (2.f * xipv[lane * 8 + j] - xip2[lane * 8 + j])
                                   : xipv[lane * 8 + j];
                xiw[lane * 8 + j] = x0;
                xs[lane * LDX + j] = x0;
            }
        }
        float errv = 3.0e38f;
        int cnt = 0;

        for (int it = 0; it < NITER; ++it) {
            bool active = (errv > TOLF) && (cnt < NITER);   // batch-global scalar
            if (!active) break;

            // A: pre_u = x@Wu0 + bu0 -> u1 (Ub), sigmoid (Sb)
            for (int nt = 0; nt < 4; ++nt) {
                v8f c0 = mm16g(40, lane,
                    [&](int m, int k) { return xs[m * LDX + k]; },
                    [&](int k, int n) { return Wu0g[k * 64 + nt * 16 + n]; }, vzero());
                st16(c0, lane, [&](int r, int n2, float v) {
                    int n = nt * 16 + n2;
                    float pre = v + bu0v[n];
                    Ub[r * LD64 + n] = sp_(pre);
                    Sb[r * LD64 + n] = sig_(pre);
                });
            }
            // B: z1pre = x@Wx1 + b1 -> Ab
            for (int nt = 0; nt < 4; ++nt) {
                v8f c0 = mm16g(40, lane,
                    [&](int m, int k) { return xs[m * LDX + k]; },
                    [&](int k, int n) { return Wx1g[k * 64 + nt * 16 + n]; }, vzero());
                st16(c0, lane, [&](int r, int n2, float v) {
                    int n = nt * 16 + n2;
                    Ab[r * LD64 + n] = v + b1v[n];
                });
            }
            // C: t = x@Wyx1 + byx1 (per item)
            if (lane < 16) {
                float s = byx1s;
                for (int i = 0; i < 40; ++i) s += xs[lane * LDX + i] * Wyx1v[i];
                t_s[lane] = s;
            }
            // D: z1pre += (y*t)*Wy1
            for (int i = lane; i < 16 * 64; i += 32) {
                int m = i >> 6, k = i & 63;
                Ab[m * LD64 + k] += y_s[m] * t_s[m] * Wy1row[k];
            }
            // E: zg = relu(u1@Wzu2 + bzu2) -> Bb
            for (int nt = 0; nt < 4; ++nt) {
                v8f c0 = mm16g(64, lane,
                    [&](int m, int k) { return Ub[m * LD64 + k]; },
                    [&](int k, int n) { return Wzu2s[k * LD64 + nt * 16 + n]; }, vzero());
                st16(c0, lane, [&](int r, int n2, float v) {
                    int n = nt * 16 + n2;
                    float zp = v + bzu2v[n];
                    Bb[r * LD64 + n] = (zp > 0.f) ? zp : 0.f;
                });
            }
            // F: dz = sig(z1pre)*zg*wz (Ab), dzg = sp(z1pre)*wz*[zg>0] (Bb)
            for (int i = lane; i < 16 * 64; i += 32) {
                int m = i >> 6, k = i & 63;
                float zp = Ab[m * LD64 + k];
                float zg = Bb[m * LD64 + k];
                float w = wzv[k];
                Ab[m * LD64 + k] = sig_(zp) * zg * w;
                Bb[m * LD64 + k] = (zg > 0.f) ? sp_(zp) * w : 0.f;
            }
            // G: yg, S = dz . Wy1 ; gu = t*S + yg*Wy2
            {
                int m = lane & 15;
                if (lane < 16) {
                    float s = byu2s;
                    for (int k = 0; k < 64; ++k) s += Ub[m * LD64 + k] * Wyu2v[k];
                    yg_s[m] = s;
                } else {
                    float s = 0.f;
                    for (int k = 0; k < 64; ++k) s += Ab[m * LD64 + k] * Wy1row[k];
                    S_s[m] = s;
                }
                if (lane < 16) gu_s[m] = t_s[m] * S_s[m] + yg_s[m] * Wy2s;
            }
            // H: h1 = sig(u) * (dzg@Wzu2^T + y*Wy2*Wyu2 + Wu2) -> Ub
            for (int nt = 0; nt < 4; ++nt) {
                v8f c0 = mm16g(64, lane,
                    [&](int m, int k) { return Bb[m * LD64 + k]; },
                    [&](int k, int n) { return Wzu2s[(nt * 16 + n) * LD64 + k]; }, vzero());
                st16(c0, lane, [&](int r, int n2, float v) {
                    int n = nt * 16 + n2;
                    float d = v + y_s[r] * Wy2s * Wyu2v[n] + Wu2v[n];
                    Ub[r * LD64 + n] = Sb[r * LD64 + n] * d;
                });
            }
            // I: gxa = h1 @ Wu0^T  (only first 8 x-components needed)
            {
                v8f c0 = mm16g(64, lane,
                    [&](int m, int k) { return Ub[m * LD64 + k]; },
                    [&](int k, int n) { return Wu0g[n * 64 + k]; }, vzero());
                st16(c0, lane, [&](int r, int n2, float v) { gxa[r * 17 + n2] = v; });
            }
            // J: gxb = dz @ Wx1^T
            {
                v8f c0 = mm16g(64, lane,
                    [&](int m, int k) { return Ab[m * LD64 + k]; },
                    [&](int k, int n) { return Wx1g[n * 64 + k]; }, vzero());
                st16(c0, lane, [&](int r, int n2, float v) { gxb[r * 17 + n2] = v; });
            }
            // K: dissipation: es = sig((q*a_x)@Wy1d + wxc) * gk -> Sb
            if (lane < 16) {
#pragma unroll
                for (int j = 0; j < 8; ++j) {
                    float q = (xiw[lane * 8 + j] - xipv[lane * 8 + j]) * DT_INV;
                    sqs[lane * 12 + j] = q * Axs[lane * 8 + j];
                }
#pragma unroll
                for (int j = 8; j < 12; ++j) sqs[lane * 12 + j] = 0.f;
            }
            for (int nt = 0; nt < 4; ++nt) {
                v8f c0 = mm16g(8, lane,
                    [&](int m, int k) { return sqs[m * 12 + k]; },
                    [&](int k, int n) { return Wy1ds[k * LD64 + nt * 16 + n]; }, vzero());
                st16(c0, lane, [&](int r, int n2, float v) {
                    int n = nt * 16 + n2;
                    float sv = v + wxcs[r * 64 + n];
                    Sb[r * LD64 + n] = sig_(sv) * gks[r * 64 + n];
                });
            }
            // L: ddx = es @ Wy1d^T (rows >=8 of Wy1ds zeroed)
            {
                v8f c0 = mm16g(64, lane,
                    [&](int m, int k) { return Sb[m * LD64 + k]; },
                    [&](int k, int n) { return Wy1ds[n * LD64 + k]; }, vzero());
                st16(c0, lane, [&](int r, int n2, float v) { ddx[r * 17 + n2] = v; });
            }
            // M: g = gv + dD/dq, commit xi/s_eq (active is uniformly true here)
            if (lane < 16) {
                int m = lane;
                float gsum = 0.f;
                float gj[8];
#pragma unroll
                for (int j = 0; j < 8; ++j) {
                    float g = gxa[m * 17 + j] + gxb[m * 17 + j]
                            + y_s[m] * S_s[m] * Wyx1v[j]
                            + Axs[m * 8 + j] * ddx[m * 17 + j] + rjs[m * 8 + j];
                    gj[j] = g;
                    gsum += fabsf(g);
                }
                psum[m] = gsum;
#pragma unroll
                for (int j = 0; j < 8; ++j) {
                    float nv = xiw[m * 8 + j] - LRF * gj[j];
                    xiw[m * 8 + j] = nv;
                    xs[m * LDX + j] = nv;
                }
                seq_s[m] = gu_s[m];
            }
            // global deterministic err reduction + slot barrier
            if (lane == 0) {
                float bs = 0.f;
                for (int m2 = 0; m2 < 16; ++m2) bs += psum[m2];
                int slot = t * NITER + it;
                unsigned long long q =
                    (unsigned long long)((double)bs * 16777216.0);   // Q24
                __hip_atomic_fetch_add(&acc[slot], q, __ATOMIC_RELAXED,
                                       __HIP_MEMORY_SCOPE_AGENT);
                __hip_atomic_fetch_add(&bar[slot], 1u, __ATOMIC_ACQ_REL,
                                       __HIP_MEMORY_SCOPE_AGENT);
                while (__hip_atomic_load(&bar[slot], __ATOMIC_ACQUIRE,
                                         __HIP_MEMORY_SCOPE_AGENT) < (unsigned)gridDim.x) {
                    __builtin_amdgcn_s_sleep(2);
                }
                unsigned long long tot = __hip_atomic_load(
                    &acc[slot], __ATOMIC_ACQUIRE, __HIP_MEMORY_SCOPE_AGENT);
                errsh = (float)((double)tot * (1.0 / 16777216.0)
                                * (1.0 / ((double)NB * (double)NIVD)));
            }
            errv = errsh;
            cnt += 1;
        }

        // outputs: stress = last-active gu, xi output = pre-step xi
        if (lane < 16) {
            int m = lane;
            stress[(base + m) * NT + t] = seq_s[m];
#pragma unroll
            for (int j = 0; j < 8; ++j) {
                xiout[((base + m) * NT + t) * 8 + j] = xipv[m * 8 + j];
                xip2[m * 8 + j] = xipv[m * 8 + j];
                xipv[m * 8 + j] = xiw[m * 8 + j];
            }
        }
    }
}

// ===========================================================================
extern "C" void kernel_launch(void* const* d_in, const int* in_sizes, int n_in,
                              void* d_out, int out_size, void* d_ws, size_t ws_size,
                              hipStream_t stream) {
    (void)in_sizes; (void)n_in; (void)out_size; (void)ws_size;
    auto F = [&](int i) { return (const float*)d_in[i]; };
    // Top level in setup_inputs() insertion order; nested dicts in sorted-key
    // (pytree) order.
    const float* e  = F(0);
    const float* E  = F(1);
    const float* Yv = F(2);
    const float* nv = F(3);
    const float* ed = F(4);
    PicnnP ep{F(5),  F(6),  F(7),  F(8),  F(9),  F(10), F(11), F(12), F(13),
              F(14), F(15), F(16), F(17), F(18), F(19)};
    PicnnP dp{F(20), F(21), F(22), F(23), F(24), F(25), F(26), F(27), F(28),
              F(29), F(30), F(31), F(32), F(33), F(34)};
    FnoP f1{F(35), F(36), F(37), F(38), F(39), F(40), F(41), F(42), F(43),
            F(44), F(45), F(46), F(47), F(48), F(49), F(50), F(51), F(52)};
    FnoP f2{F(53), F(54), F(55), F(56), F(57), F(58), F(59), F(60), F(61),
            F(62), F(63), F(64), F(65), F(66), F(67), F(68), F(69), F(70)};

    float* w    = (float*)d_ws;
    float* mf1  = w;
    float* mf2  = mf1 + NB * 32;
    float* Axw  = mf2 + NB * 32;
    float* gkw  = Axw + NB * 8;
    float* wxw  = gkw + NB * 64;
    float* rjw  = wxw + NB * 64;
    float* endf = rjw + NB * 8;
    size_t off = ((size_t)((char*)endf - (char*)d_ws) + 7) & ~(size_t)7;
    unsigned long long* acc = (unsigned long long*)((char*)d_ws + off);
    unsigned* bar = (unsigned*)(acc + NT * NITER);
    hipMemsetAsync(acc, 0,
                   NT * NITER * (sizeof(unsigned long long) + sizeof(unsigned)),
                   stream);

    fno_kernel<<<NB, 64, 0, stream>>>(E, nullptr, nullptr, 1, f1, mf1);
    fno_kernel<<<NB, 64, 0, stream>>>(Yv, nv, ed, 3, f2, mf2);
    dd_precompute_kernel<<<NB / 128, 128, 0, stream>>>(mf2, dp, Axw, gkw, wxw, rjw);

    float* stress = (float*)d_out;
    float* xiout  = stress + NB * NT;
    time_loop_kernel<<<NB / 16, 32, 0, stream>>>(e, mf1, ep, dp.Wy1, Axw, gkw,
                                                 wxw, rjw, acc, bar, stress, xiout);
}